// EXPC_29970281791594
// MI455X (gfx1250) — compile-verified
//
#include <hip/hip_runtime.h>

// ---------------------------------------------------------------------------
// Problem constants (match the reference)
// ---------------------------------------------------------------------------
#define N_NODES 50000
#define N_EDGES 800000
#define N_GRAPH 512
#define HDIM    128
#define EFDIM   16
#define LLAYERS 3
#define BN_EPS  1e-5f

typedef float v2f __attribute__((ext_vector_type(2)));
typedef float v8f __attribute__((ext_vector_type(8)));

// ---------------------------------------------------------------------------
// Async stage of one 16 x Nc k-chunk of W into LDS (double-buffered by c&1).
// Uses the CDNA5 async-to-LDS engine (ASYNCcnt); every thread issues b128
// copies for its share of the chunk.  Generic pointers to __shared__ carry the
// LDS byte offset in their low 32 bits (ISA 10.2), so the truncating cast
// yields a valid LDS VDST address.
// ---------------------------------------------------------------------------
__device__ __forceinline__ void stage_chunk_async(const float* __restrict__ W,
                                                  float* sW, int c, int chunkFloats)
{
    const float* gsrc  = W + (size_t)c * chunkFloats;
    float*       lbase = sW + (c & 1) * chunkFloats;
    const int nf4 = chunkFloats >> 2;
    for (int idx = threadIdx.x; idx < nf4; idx += blockDim.x) {
        unsigned           lds = (unsigned)(size_t)(lbase + idx * 4);
        unsigned long long ga  = (unsigned long long)(size_t)(gsrc + idx * 4);
        asm volatile("global_load_async_to_lds_b128 %0, %1, off"
                     :: "v"(lds), "v"(ga) : "memory");
    }
}

// ---------------------------------------------------------------------------
// fp32 WMMA GEMM with LDS-pipelined W:
//   C[M,Nc] = relu?( (A (+A2)) @ W + bias ),  M = Mtiles*16 rows processed
// - wave computes a 16(M) x 128(N) strip: 8 accumulator tiles reuse one A frag
// - W streamed in 16 x Nc chunks, double-buffered in LDS via async-to-LDS,
//   overlapped with the 32 WMMAs of the previous chunk
// - block covers the full Nc extent of its row strips; __syncthreads() before
//   the store phase makes in-place C (aliasing A or A2) safe
// - tail waves (mtile >= Mtiles) skip compute wave-uniformly (EXEC all-ones
//   for WMMA) but still stage + hit barriers
// ---------------------------------------------------------------------------
__global__ void wmma_gemm_lds(const float* __restrict__ A,
                              const float* __restrict__ A2,   // optional
                              const float* __restrict__ W,    // [K, Nc] row-major
                              const float* __restrict__ bias, // [Nc]
                              float* __restrict__ C,
                              int Mtiles, int K, int Nc, int do_relu)
{
    extern __shared__ float sW[];                 // 2 * 16 * Nc floats
    const int waves = blockDim.x >> 5;
    const int Cg    = Nc >> 7;                    // column groups of 128
    const int R     = waves / Cg;                 // row strips per block
    const int lane  = threadIdx.x & 31;
    const int w     = threadIdx.x >> 5;
    const int rstrip = w / Cg;
    const int cgrp   = w % Cg;
    const int half = lane >> 4;
    const int lrow = lane & 15;
    const int mtile = blockIdx.x * R + rstrip;
    const bool active = (mtile < Mtiles);         // wave-uniform
    const int ncol0 = cgrp * 128;
    const int chunkFloats = 16 * Nc;
    const int nchunks = K >> 4;

    stage_chunk_async(W, sW, 0, chunkFloats);     // prologue

    const float* Arow  = A + (size_t)(mtile * 16 + lrow) * K;
    const float* A2row = A2 ? (A2 + (size_t)(mtile * 16 + lrow) * K) : nullptr;

    v8f acc[8] = {};
    for (int c = 0; c < nchunks; ++c) {
        asm volatile("s_wait_asynccnt 0x0" ::: "memory");
        __syncthreads();                          // chunk c resident in LDS
        if (c + 1 < nchunks)
            stage_chunk_async(W, sW, c + 1, chunkFloats);  // overlap next chunk

        if (active) {
            const float* lw = sW + (c & 1) * chunkFloats;
#pragma unroll
            for (int ks = 0; ks < 4; ++ks) {
                const int kk = ks * 4 + 2 * half;   // k within chunk (this half)
                const int kg = c * 16 + kk;         // global k
                v2f a;
                a.x = Arow[kg];
                a.y = Arow[kg + 1];
                if (A2) {                           // uniform branch
                    a.x += A2row[kg];
                    a.y += A2row[kg + 1];
                }
#pragma unroll
                for (int nt = 0; nt < 8; ++nt) {
                    const int col = ncol0 + nt * 16 + lrow;
                    v2f b;
                    b.x = lw[kk * Nc + col];
                    b.y = lw[(kk + 1) * Nc + col];
                    acc[nt] = __builtin_amdgcn_wmma_f32_16x16x4_f32(
                                  false, a, false, b, (short)0, acc[nt],
                                  false, false);
                }
            }
        }
    }

    __syncthreads();   // all reads of A/A2 rows done -> in-place store safe
    if (active) {
#pragma unroll
        for (int nt = 0; nt < 8; ++nt) {
            const int col = ncol0 + nt * 16 + lrow;
            const float bv = bias ? bias[col] : 0.0f;
#pragma unroll
            for (int v = 0; v < 8; ++v) {
                const int m = mtile * 16 + v + 8 * half;
                float val = acc[nt][v] + bv;
                if (do_relu) val = fmaxf(val, 0.0f);
                C[(size_t)m * Nc + col] = val;
            }
        }
    }
}

// ---------------------------------------------------------------------------
// Fused edge kernel (warp per edge):
//   msg = relu( h[src] + edge_attr @ We + be );  agg[dst] += msg
// We (16x128, 8KB) and be live in LDS; lane handles 4 stride-32 columns ->
// conflict-free LDS banks, coalesced h gathers and agg atomics (L2-resident).
// ---------------------------------------------------------------------------
#define EDGES_PER_BLOCK 128
__global__ void edge_msg_kernel(const float* __restrict__ h,
                                const float* __restrict__ edge_attr, // [E,16]
                                const int*   __restrict__ ei,        // [2,E]
                                const float* __restrict__ We,        // [16,128]
                                const float* __restrict__ be,        // [128]
                                float* __restrict__ agg, int E)
{
    __shared__ float sWe[EFDIM * HDIM];
    __shared__ float sbe[HDIM];
    for (int i = threadIdx.x; i < EFDIM * HDIM; i += blockDim.x) sWe[i] = We[i];
    if (threadIdx.x < HDIM) sbe[threadIdx.x] = be[threadIdx.x];
    __syncthreads();

    const int lane = threadIdx.x & 31;
    const int wid  = threadIdx.x >> 5;       // 8 warps
    const int base = blockIdx.x * EDGES_PER_BLOCK;
    const int* src = ei;
    const int* dst = ei + E;

    for (int it = 0; it < EDGES_PER_BLOCK / 8; ++it) {
        const int e = base + wid * (EDGES_PER_BLOCK / 8) + it;
        if (e >= E) break;
        if (e + EDGES_PER_BLOCK < E)   // stream-ahead hint -> global_prefetch_b8
            __builtin_prefetch(edge_attr + (size_t)(e + EDGES_PER_BLOCK) * EFDIM, 0, 0);

        float eav[EFDIM];
        const float* eap = edge_attr + (size_t)e * EFDIM;
#pragma unroll
        for (int k = 0; k < EFDIM; ++k) eav[k] = eap[k];

        const int s = src[e];
        const int d = dst[e];
        const float* hrow = h   + (size_t)s * HDIM;
        float*       arow = agg + (size_t)d * HDIM;

#pragma unroll
        for (int c = 0; c < 4; ++c) {
            const int col = c * 32 + lane;
            float acc = sbe[col];
#pragma unroll
            for (int k = 0; k < EFDIM; ++k)
                acc = fmaf(eav[k], sWe[k * HDIM + col], acc);
            float m = hrow[col] + acc;
            m = fmaxf(m, 0.0f);
            atomicAdd(&arow[col], m);
        }
    }
}

// ---------------------------------------------------------------------------
// Small elementwise / reduction helpers
// ---------------------------------------------------------------------------
__global__ void zero_f(float* p, int n)
{
    int i = blockIdx.x * blockDim.x + threadIdx.x;
    if (i < n) p[i] = 0.0f;
}

__global__ void broadcast_vn(float* vn, const float* vn_emb, int n)
{
    int i = blockIdx.x * blockDim.x + threadIdx.x;
    if (i < n) vn[i] = vn_emb[i & (HDIM - 1)];
}

// h += vn[batch]; agg = 0
__global__ void add_vn_zero(float* __restrict__ h, const float* __restrict__ vn,
                            const int* __restrict__ batch, float* __restrict__ agg, int n)
{
    int i = blockIdx.x * blockDim.x + threadIdx.x;
    if (i >= n) return;
    const int node = i >> 7, j = i & (HDIM - 1);
    h[i] += vn[(size_t)batch[node] * HDIM + j];
    agg[i] = 0.0f;
}

// per-column sum & sum-of-squares of X[M,C] -> stats[0..C)=sum, [C..2C)=sumsq
__global__ void col_stats(const float* __restrict__ X, float* __restrict__ stats,
                          int M, int C, int rows_per_block)
{
    const int col     = threadIdx.x % C;
    const int rOff    = threadIdx.x / C;
    const int rStride = blockDim.x / C;
    const int r0 = blockIdx.x * rows_per_block;
    const int r1 = min(r0 + rows_per_block, M);
    float s = 0.0f, sq = 0.0f;
    for (int r = r0 + rOff; r < r1; r += rStride) {
        const float v = X[(size_t)r * C + col];
        s += v;
        sq = fmaf(v, v, sq);
    }
    atomicAdd(&stats[col], s);
    atomicAdd(&stats[C + col], sq);
}

// h = relu(bn(tmp));  t[:,0:H] = vn[batch];  t[:,H:2H] = h
__global__ void bn_relu_concat(const float* __restrict__ tmp, const float* __restrict__ stats,
                               const float* __restrict__ g, const float* __restrict__ beta,
                               const float* __restrict__ vn, const int* __restrict__ batch,
                               float* __restrict__ h, float* __restrict__ t, int n, float invM)
{
    int i = blockIdx.x * blockDim.x + threadIdx.x;
    if (i >= n) return;
    const int node = i >> 7, j = i & (HDIM - 1);
    const float mu  = stats[j] * invM;
    const float var = stats[HDIM + j] * invM - mu * mu;
    float y = g[j] * (tmp[i] - mu) * rsqrtf(var + BN_EPS) + beta[j];
    y = fmaxf(y, 0.0f);
    h[i] = y;
    t[(size_t)node * (2 * HDIM) + HDIM + j] = y;
    t[(size_t)node * (2 * HDIM) + j]        = vn[(size_t)batch[node] * HDIM + j];
}

// y = relu(bn(X[N,2H]));  pooled[batch[n]] += y   (segment_sum fused)
__global__ void bn_relu_pool(const float* __restrict__ X, const float* __restrict__ stats,
                             const float* __restrict__ g, const float* __restrict__ beta,
                             const int* __restrict__ batch, float* __restrict__ pooled,
                             int n, float invM)
{
    int i = blockIdx.x * blockDim.x + threadIdx.x;
    if (i >= n) return;
    const int node = i >> 8, j = i & (2 * HDIM - 1);
    const float mu  = stats[j] * invM;
    const float var = stats[2 * HDIM + j] * invM - mu * mu;
    float y = g[j] * (X[i] - mu) * rsqrtf(var + BN_EPS) + beta[j];
    y = fmaxf(y, 0.0f);
    atomicAdd(&pooled[(size_t)batch[node] * (2 * HDIM) + j], y);
}

// vn = relu(bn(X[G,H]));  hg[:, layer*H : (layer+1)*H] = vn   (JK concat)
__global__ void bn_relu_vn(const float* __restrict__ X, const float* __restrict__ stats,
                           const float* __restrict__ g, const float* __restrict__ beta,
                           float* __restrict__ vn, float* __restrict__ hg,
                           int n, int layer, float invM)
{
    int i = blockIdx.x * blockDim.x + threadIdx.x;
    if (i >= n) return;
    const int gi = i >> 7, j = i & (HDIM - 1);
    const float mu  = stats[j] * invM;
    const float var = stats[HDIM + j] * invM - mu * mu;
    float y = g[j] * (X[i] - mu) * rsqrtf(var + BN_EPS) + beta[j];
    y = fmaxf(y, 0.0f);
    vn[i] = y;
    hg[(size_t)gi * (LLAYERS * HDIM) + layer * HDIM + j] = y;
}

// out[g] = dot(pg[g,:384], w2) + b2   (warp per graph)
__global__ void pred_out(const float* __restrict__ pg, const float* __restrict__ w2,
                         const float* __restrict__ b2, float* __restrict__ out, int G)
{
    const int lane = threadIdx.x & 31;
    const int warp = threadIdx.x >> 5;
    const int gi = blockIdx.x * (blockDim.x >> 5) + warp;
    if (gi >= G) return;
    const float* row = pg + (size_t)gi * (LLAYERS * HDIM);
    float s = 0.0f;
    for (int j = lane; j < LLAYERS * HDIM; j += 32) s = fmaf(row[j], w2[j], s);
#pragma unroll
    for (int off = 16; off > 0; off >>= 1) s += __shfl_down(s, off, 32);
    if (lane == 0) out[gi] = s + b2[0];
}

// ---------------------------------------------------------------------------
// Host orchestration
// ---------------------------------------------------------------------------
extern "C" void kernel_launch(void* const* d_in, const int* in_sizes, int n_in,
                              void* d_out, int out_size, void* d_ws, size_t ws_size,
                              hipStream_t stream)
{
    const float* x         = (const float*)d_in[0];
    const float* edge_attr = (const float*)d_in[1];
    const float* atom_W    = (const float*)d_in[2];
    const float* atom_b    = (const float*)d_in[3];
    const float* vn_emb    = (const float*)d_in[4];
    const float* conv_We   = (const float*)d_in[5];
    const float* conv_be   = (const float*)d_in[6];
    const float* conv_W    = (const float*)d_in[7];
    const float* conv_b    = (const float*)d_in[8];
    const float* conv_g    = (const float*)d_in[9];
    const float* conv_beta = (const float*)d_in[10];
    const float* vn1_W     = (const float*)d_in[11];
    const float* vn1_b     = (const float*)d_in[12];
    const float* vn1_g     = (const float*)d_in[13];
    const float* vn1_beta  = (const float*)d_in[14];
    const float* vn2_W     = (const float*)d_in[15];
    const float* vn2_b     = (const float*)d_in[16];
    const float* vn2_g     = (const float*)d_in[17];
    const float* vn2_beta  = (const float*)d_in[18];
    const float* pred_W1   = (const float*)d_in[19];
    const float* pred_b1   = (const float*)d_in[20];
    const float* pred_W2   = (const float*)d_in[21];
    const float* pred_b2   = (const float*)d_in[22];
    const int*   edge_idx  = (const int*)d_in[23];
    const int*   batch     = (const int*)d_in[24];
    float*       out       = (float*)d_out;

    const int N = N_NODES, E = N_EDGES, G = N_GRAPH, H = HDIM;
    const size_t NH = (size_t)N * H, N2H = (size_t)N * 2 * H;

    // workspace layout (floats)
    float* ws      = (float*)d_ws;
    float* h       = ws;                       // [N,H]
    float* agg     = h + NH;                   // [N,H]   (also node-GEMM output)
    float* t       = agg + NH;                 // [N,2H]  (vn1 GEMM runs in-place)
    float* vn      = t + N2H;                  // [G,H]
    float* pooled  = vn + (size_t)G * H;       // [G,2H]
    float* pooled2 = pooled + (size_t)G * 2*H; // [G,H]
    float* hg      = pooled2 + (size_t)G * H;  // [G,3H]
    float* pg      = hg + (size_t)G * 3 * H;   // [G,3H]
    float* stats   = pg + (size_t)G * 3 * H;   // [512]

    const int TPB = 256;
    const int MtN = N / 16;   // 3125 (exact)
    const int MtG = G / 16;   // 32

    // GEMM launch configs: block covers R row strips x full Nc; R = waves/Cg
    const int shm128 = 2 * 16 * 128 * sizeof(float);   // 16 KB
    const int shm256 = 2 * 16 * 256 * sizeof(float);   // 32 KB
    const int shm384 = 2 * 16 * 384 * sizeof(float);   // 48 KB
    const int gN_R8 = (MtN + 7) / 8;    // Nc=128: 256 thr, R=8
    const int gN_R4 = (MtN + 3) / 4;    // Nc=256: 256 thr, R=4

    // atom encoder: h = x @ atom_W + atom_b
    wmma_gemm_lds<<<gN_R8, 256, shm128, stream>>>(x, nullptr, atom_W, atom_b, h,
                                                  MtN, H, H, 0);
    // vn = broadcast(vn_emb)
    broadcast_vn<<<(G*H + TPB-1)/TPB, TPB, 0, stream>>>(vn, vn_emb, G*H);

    for (int i = 0; i < LLAYERS; ++i) {
        // h += vn[batch]; agg = 0
        add_vn_zero<<<(int)((NH + TPB-1)/TPB), TPB, 0, stream>>>(h, vn, batch, agg, (int)NH);
        // agg += relu(h[src] + edge_attr@We + be) scattered by dst
        edge_msg_kernel<<<(E + EDGES_PER_BLOCK-1)/EDGES_PER_BLOCK, TPB, 0, stream>>>(
            h, edge_attr, edge_idx, conv_We + (size_t)i*EFDIM*H, conv_be + (size_t)i*H, agg, E);
        // agg <- (h + agg) @ conv_W + conv_b   (in-place over agg; safe, see kernel)
        wmma_gemm_lds<<<gN_R8, 256, shm128, stream>>>(
            h, agg, conv_W + (size_t)i*H*H, conv_b + (size_t)i*H, agg, MtN, H, H, 0);
        // BN stats over N rows, C=H
        zero_f<<<1, TPB, 0, stream>>>(stats, 2*H);
        col_stats<<<(N + 63)/64, TPB, 0, stream>>>(agg, stats, N, H, 64);
        // h = relu(bn(agg)); t = [vn[batch], h]
        bn_relu_concat<<<(int)((NH + TPB-1)/TPB), TPB, 0, stream>>>(
            agg, stats, conv_g + (size_t)i*H, conv_beta + (size_t)i*H, vn, batch, h, t,
            (int)NH, 1.0f/(float)N);
        // t <- t @ vn1_W + vn1_b (in-place)
        wmma_gemm_lds<<<gN_R4, 256, shm256, stream>>>(
            t, nullptr, vn1_W + (size_t)i*2*H*2*H, vn1_b + (size_t)i*2*H, t, MtN, 2*H, 2*H, 0);
        // BN stats over N rows, C=2H; pooled = segment_sum(relu(bn(t)), batch)
        zero_f<<<2, TPB, 0, stream>>>(stats, 4*H);
        col_stats<<<(N + 63)/64, TPB, 0, stream>>>(t, stats, N, 2*H, 64);
        zero_f<<<(G*2*H + TPB-1)/TPB, TPB, 0, stream>>>(pooled, G*2*H);
        bn_relu_pool<<<(int)((N2H + TPB-1)/TPB), TPB, 0, stream>>>(
            t, stats, vn1_g + (size_t)i*2*H, vn1_beta + (size_t)i*2*H, batch, pooled,
            (int)N2H, 1.0f/(float)N);
        // pooled2 = pooled @ vn2_W + vn2_b
        wmma_gemm_lds<<<(MtG + 7)/8, 256, shm128, stream>>>(
            pooled, nullptr, vn2_W + (size_t)i*2*H*H, vn2_b + (size_t)i*H, pooled2,
            MtG, 2*H, H, 0);
        // BN over G rows; vn = relu(bn); hg[:, i*H:(i+1)*H] = vn
        zero_f<<<1, TPB, 0, stream>>>(stats, 2*H);
        col_stats<<<(G + 63)/64, TPB, 0, stream>>>(pooled2, stats, G, H, 64);
        bn_relu_vn<<<(G*H + TPB-1)/TPB, TPB, 0, stream>>>(
            pooled2, stats, vn2_g + (size_t)i*H, vn2_beta + (size_t)i*H, vn, hg,
            G*H, i, 1.0f/(float)G);
    }

    // prediction head: pg = relu(hg @ pred_W1 + pred_b1)   (384 thr: Cg=3, R=4)
    wmma_gemm_lds<<<(MtG + 3)/4, 384, shm384, stream>>>(
        hg, nullptr, pred_W1, pred_b1, pg, MtG, 3*H, 3*H, 1);
    pred_out<<<G/8, TPB, 0, stream>>>(pg, pred_W2, pred_b2, out, G);
}